// NeuralODE_18485539242014
// MI455X (gfx1250) — compile-verified
//
#include <hip/hip_runtime.h>
#include <hip/hip_bf16.h>

typedef float v2f __attribute__((ext_vector_type(2)));
typedef float v8f __attribute__((ext_vector_type(8)));

#define RGAS   8.314f
#define A_0    1.0e10f
#define H_COEF 20.0f
#define RHO    700.0f
#define CP     2000.0f
#define Y_C    0.35f
#define Y_H    0.60f
#define Y_L    0.75f

__device__ __forceinline__ float sigmoidf(float x) {
    return 1.0f / (1.0f + __expf(-x));
}

// ---------------------------------------------------------------------------
// Kernel 1: MLP  (X[B,2] -> EA[B,10])  using V_WMMA_F32_16X16X4_F32
// One wave (32 lanes) processes 16 batch rows. 8 waves / block.
// ---------------------------------------------------------------------------
#define LDSTRIDE 66   // 16 rows x 64 cols, stride 66 (even => v2f aligned, low bank conflicts)

__global__ __launch_bounds__(256) void mlp_ea_kernel(
    const float* __restrict__ X,
    const float* __restrict__ W0, const float* __restrict__ b0,
    const float* __restrict__ W1, const float* __restrict__ b1,
    const float* __restrict__ W2, const float* __restrict__ b2,
    float* __restrict__ EA, int B)
{
    __shared__ float lds[8][16 * LDSTRIDE];

    const int wave = threadIdx.x >> 5;
    const int lane = threadIdx.x & 31;
    const int half = lane >> 4;     // 0: lanes 0-15, 1: lanes 16-31
    const int l16  = lane & 15;
    const int rowBase = (blockIdx.x * 8 + wave) * 16;
    float* tile = lds[wave];

    // ---- layer 0:  xn[2] -> h0[64], sigmoid.  Each lane: 32 outputs of row l16.
    {
        int row = rowBase + l16;
        int rc  = row < B ? row : (B - 1);
        float xn0 = (X[rc * 2 + 0] - 0.02f ) * (1.0f / 0.01f );
        float xn1 = (X[rc * 2 + 1] - 0.004f) * (1.0f / 0.002f);
        #pragma unroll
        for (int jj = 0; jj < 32; ++jj) {
            int j = half * 32 + jj;
            float z = xn0 * W0[j * 2 + 0] + xn1 * W0[j * 2 + 1] + b0[j];
            tile[l16 * LDSTRIDE + j] = sigmoidf(z);
        }
    }
    __syncthreads();

    // ---- layer 1:  h0[16x64] @ W1^T[64x64] via 64x WMMA f32 16x16x4
    v8f acc0 = {0,0,0,0,0,0,0,0};
    v8f acc1 = acc0, acc2 = acc0, acc3 = acc0;
    #pragma unroll
    for (int kb = 0; kb < 16; ++kb) {
        const int col = kb * 4 + half * 2;   // even => 8B aligned
        v2f a  = *(const v2f*)&tile[l16 * LDSTRIDE + col];          // A[M=l16][K=col..col+1]
        v2f bf0 = *(const v2f*)&W1[( 0 + l16) * 64 + col];          // B[K][N]=W1[N][K]
        v2f bf1 = *(const v2f*)&W1[(16 + l16) * 64 + col];
        v2f bf2 = *(const v2f*)&W1[(32 + l16) * 64 + col];
        v2f bf3 = *(const v2f*)&W1[(48 + l16) * 64 + col];
        acc0 = __builtin_amdgcn_wmma_f32_16x16x4_f32(false, a, false, bf0, (short)0, acc0, false, false);
        acc1 = __builtin_amdgcn_wmma_f32_16x16x4_f32(false, a, false, bf1, (short)0, acc1, false, false);
        acc2 = __builtin_amdgcn_wmma_f32_16x16x4_f32(false, a, false, bf2, (short)0, acc2, false, false);
        acc3 = __builtin_amdgcn_wmma_f32_16x16x4_f32(false, a, false, bf3, (short)0, acc3, false, false);
    }
    __syncthreads();

    // ---- bias + sigmoid, write h1 back to LDS (D layout: vgpr v -> M=v+8*half, N=16*nt+l16)
    {
        float bb0 = b1[ 0 + l16], bb1 = b1[16 + l16];
        float bb2 = b1[32 + l16], bb3 = b1[48 + l16];
        #pragma unroll
        for (int v = 0; v < 8; ++v) {
            int M = v + 8 * half;
            tile[M * LDSTRIDE +  0 + l16] = sigmoidf(acc0[v] + bb0);
            tile[M * LDSTRIDE + 16 + l16] = sigmoidf(acc1[v] + bb1);
            tile[M * LDSTRIDE + 32 + l16] = sigmoidf(acc2[v] + bb2);
            tile[M * LDSTRIDE + 48 + l16] = sigmoidf(acc3[v] + bb3);
        }
    }
    __syncthreads();

    // ---- layer 2:  h1[16x64] @ W2^T[64x10] (N padded to 16) via 16x WMMA
    v8f accE = {0,0,0,0,0,0,0,0};
    #pragma unroll
    for (int kb = 0; kb < 16; ++kb) {
        const int col = kb * 4 + half * 2;
        v2f a  = *(const v2f*)&tile[l16 * LDSTRIDE + col];
        v2f bf = {0.0f, 0.0f};
        if (l16 < 10) bf = *(const v2f*)&W2[l16 * 64 + col];
        accE = __builtin_amdgcn_wmma_f32_16x16x4_f32(false, a, false, bf, (short)0, accE, false, false);
    }
    {
        float b2v = (l16 < 10) ? b2[l16] : 0.0f;
        #pragma unroll
        for (int v = 0; v < 8; ++v) {
            int M   = v + 8 * half;
            int row = rowBase + M;
            if (l16 < 10 && row < B)
                EA[row * 10 + l16] = (accE[v] + b2v) * 10.0f;
        }
    }
}

// ---------------------------------------------------------------------------
// Kernel 2: RK4 chemistry integration. One thread per batch element.
// ---------------------------------------------------------------------------
__device__ __forceinline__ void chem_vf(const float* __restrict__ s,
                                        const float* __restrict__ c,   // c[j] = -EA[j]/R
                                        float alpha, float Tov,
                                        float* __restrict__ d)
{
    float invT = 1.0f / s[0];
    float k[10];
    #pragma unroll
    for (int j = 0; j < 10; ++j) k[j] = A_0 * __expf(c[j] * invT);   // v_exp_f32

    d[0] = alpha * (Tov - s[0]);
    d[1] = -k[0] * s[1];
    d[2] = -k[3] * s[2];
    d[3] = -k[6] * s[3];
    d[4] =  k[0] * s[1] - (k[1] + k[2]) * s[4];
    d[5] =  k[3] * s[2] - (k[4] + k[5]) * s[5];
    d[6] =  k[6] * s[3] - (k[7] + k[8]) * s[6];
    d[7] =  k[1] * s[4] + k[4] * s[5] + k[7] * s[6] - k[9] * s[7];
    d[8] = (1.0f - Y_C) * k[2] * s[4] + (1.0f - Y_H) * k[5] * s[5]
         + (1.0f - Y_L) * k[8] * s[6] + k[9] * s[7];
    d[9] = Y_C * k[2] * s[4] + Y_H * k[5] * s[5] + Y_L * k[8] * s[6];
}

__device__ __forceinline__ void rk4_step(float* __restrict__ s,
                                         const float* __restrict__ c,
                                         float alpha, float Tov, float dt)
{
    float k1[10], kx[10], tmp[10], acc[10];
    chem_vf(s, c, alpha, Tov, k1);
    #pragma unroll
    for (int j = 0; j < 10; ++j) { tmp[j] = s[j] + 0.5f * dt * k1[j]; acc[j] = k1[j]; }
    chem_vf(tmp, c, alpha, Tov, kx);
    #pragma unroll
    for (int j = 0; j < 10; ++j) { tmp[j] = s[j] + 0.5f * dt * kx[j]; acc[j] += 2.0f * kx[j]; }
    chem_vf(tmp, c, alpha, Tov, kx);
    #pragma unroll
    for (int j = 0; j < 10; ++j) { tmp[j] = s[j] + dt * kx[j]; acc[j] += 2.0f * kx[j]; }
    chem_vf(tmp, c, alpha, Tov, kx);
    const float dt6 = dt * (1.0f / 6.0f);
    #pragma unroll
    for (int j = 0; j < 10; ++j) s[j] += dt6 * (acc[j] + kx[j]);
}

__global__ __launch_bounds__(256) void ode_kernel(
    const float* __restrict__ ts, const float* __restrict__ X,
    const float* __restrict__ S0, const float* __restrict__ T_oven,
    const float* __restrict__ EA, float* __restrict__ out,
    int B, int T)
{
    __shared__ float tsh[64];
    if (threadIdx.x < T && threadIdx.x < 64) tsh[threadIdx.x] = ts[threadIdx.x];
    __syncthreads();

    int b = blockIdx.x * blockDim.x + threadIdx.x;
    if (b >= B) return;

    float c[10];
    #pragma unroll
    for (int j = 0; j < 10; ++j) c[j] = -EA[b * 10 + j] * (1.0f / RGAS);

    float sa  = X[b * 2 + 0];
    float vol = X[b * 2 + 1];
    float alpha = sa * H_COEF / (vol * RHO * CP);
    float Tov   = T_oven[b];

    float s[10];
    float* ob = out + (size_t)b * T * 10;
    #pragma unroll
    for (int j = 0; j < 10; ++j) { s[j] = S0[b * 10 + j]; ob[j] = s[j]; }

    for (int t = 1; t < T; ++t) {
        float dt_sub = (tsh[t] - tsh[t - 1]) * (1.0f / 8.0f);
        #pragma unroll 1
        for (int sub = 0; sub < 8; ++sub)
            rk4_step(s, c, alpha, Tov, dt_sub);
        float* o = ob + t * 10;
        #pragma unroll
        for (int j = 0; j < 10; ++j) o[j] = s[j];
    }
}

// ---------------------------------------------------------------------------
extern "C" void kernel_launch(void* const* d_in, const int* in_sizes, int n_in,
                              void* d_out, int out_size, void* d_ws, size_t ws_size,
                              hipStream_t stream) {
    const float* ts  = (const float*)d_in[0];
    const float* X   = (const float*)d_in[1];
    const float* S0  = (const float*)d_in[2];
    const float* Tov = (const float*)d_in[3];
    const float* W0  = (const float*)d_in[4];
    const float* b0  = (const float*)d_in[5];
    const float* W1  = (const float*)d_in[6];
    const float* b1  = (const float*)d_in[7];
    const float* W2  = (const float*)d_in[8];
    const float* b2  = (const float*)d_in[9];

    const int T = in_sizes[0];        // 64
    const int B = in_sizes[1] / 2;    // 32768

    float* EA = (float*)d_ws;         // [B,10] f32 scratch

    // MLP: one wave per 16 rows, 8 waves per 256-thread block
    int tiles  = (B + 15) / 16;
    int blocks = (tiles + 7) / 8;
    mlp_ea_kernel<<<blocks, 256, 0, stream>>>(X, W0, b0, W1, b1, W2, b2, EA, B);

    // ODE: one thread per batch element
    ode_kernel<<<(B + 255) / 256, 256, 0, stream>>>(ts, X, S0, Tov, EA,
                                                    (float*)d_out, B, T);
}